// InceptionDenseGCN_89816356094626
// MI455X (gfx1250) — compile-verified
//
#include <hip/hip_runtime.h>
#include <cstdint>

// InceptionDenseGCN on MI455X (gfx1250), fp32 WMMA path.
//
// Refactoring: EdgeConv msg = lrelu(self_t[dst] + nbr_t[src] - nbr_t[dst] + b)
// with self_t = x@Ws, nbr_t = x@Wn (per-node GEMMs, not per-edge), and since
// lrelu is monotonic, segment_max commutes inside it:
//   h[i] = lrelu(self_t[i] - nbr_t[i] + b + max_j nbr_t[src_j])
// => all heavy math is dense [20000 x K] @ [K x 128] GEMMs (~13 GFLOP total)
// done with V_WMMA_F32_16X16X4_F32, A/B tiles streamed to LDS with the CDNA5
// async-to-LDS path (ASYNCcnt) in a double-buffered pipeline, plus an
// L2-resident gather-max for the neighbor aggregation.

typedef __attribute__((ext_vector_type(2))) float v2f;
typedef __attribute__((ext_vector_type(8))) float v8f;
typedef int avec4 __attribute__((vector_size(16)));   // matches builtin param

#define NNODES 20000
#define KNBR   16
#define CIN    256
#define CD     128
#define SLOPE  0.2f

#if __has_builtin(__builtin_amdgcn_global_load_async_to_lds_b128) && \
    __has_builtin(__builtin_amdgcn_s_wait_asynccnt)
#define HAVE_ASYNC 1
#else
#define HAVE_ASYNC 0
#endif

#if HAVE_ASYNC
__device__ __forceinline__ void async_b128(const void* g, void* l) {
    // Builtin signature (probe-derived): 4 args, generic pointers to 16B int
    // vectors: (src, dst, imm offset, imm cpol).
    __builtin_amdgcn_global_load_async_to_lds_b128((avec4*)g, (avec4*)l, 0, 0);
}
#endif

// ---------------------------------------------------------------------------
// GEMM: Out[M x 128] = concat_k(A0|A1|A2) @ B + bias  (optional lrelu / resid)
// A segments row-major [M x ks] contiguous; B row-major [Ktot x 128].
// Block tile 64x128, 8 waves, each wave a 32x32 quadrant = 2x2 WMMA tiles.
// ---------------------------------------------------------------------------
__global__ __launch_bounds__(256)
void gemm128_wmma(const float* __restrict__ A0, int k0,
                  const float* __restrict__ A1, int k1,
                  const float* __restrict__ A2, int k2,
                  const float* __restrict__ B,
                  const float* __restrict__ bias,
                  float* __restrict__ Out, int ostride, int ooff,
                  const float* __restrict__ Res, int rstride, int roff,
                  int M, int doLrelu)
{
#if HAVE_ASYNC
    __shared__ float As[2][64][36];     // double buffered; +4 pad rows (144 B)
    __shared__ float Bs[2][32][128];
#else
    __shared__ float As[1][64][36];
    __shared__ float Bs[1][32][128];
#endif

    const int tid  = threadIdx.x;
    const int lane = tid & 31;
    const int wv   = tid >> 5;
    const int wr   = wv >> 2;              // 0..1 : row quadrant
    const int wc   = wv & 3;               // 0..3 : col quadrant
    const int l15  = lane & 15;
    const int koff = (lane >> 4) << 1;     // lanes 16-31 hold K+2,K+3
    const int row0 = blockIdx.x * 64;
    const int Ktot = k0 + k1 + k2;
    const int r    = tid >> 3;             // staging: row 0..31 (+32)
    const int q    = (tid & 7) * 4;        // staging: col (float4)

    auto segsel = [&](int kb, const float*& Ap, int& stride, int& kloc) {
        if (kb < k0)           { Ap = A0; stride = k0; kloc = kb; }
        else if (kb < k0 + k1) { Ap = A1; stride = k1; kloc = kb - k0; }
        else                   { Ap = A2; stride = k2; kloc = kb - k0 - k1; }
    };

    v8f acc00 = {}, acc01 = {}, acc10 = {}, acc11 = {};

    auto compute = [&](int bi) {
        #pragma unroll
        for (int kk = 0; kk < 32; kk += 4) {
            // A frag 16x4: lane l -> row M0+(l&15), cols kk+koff, kk+koff+1
            v2f a0 = *(const v2f*)&As[bi][wr * 32      + l15][kk + koff];
            v2f a1 = *(const v2f*)&As[bi][wr * 32 + 16 + l15][kk + koff];
            // B frag 4x16: lane l -> col N0+(l&15), rows kk+koff, kk+koff+1
            v2f b0, b1;
            b0.x = Bs[bi][kk + koff    ][wc * 32      + l15];
            b0.y = Bs[bi][kk + koff + 1][wc * 32      + l15];
            b1.x = Bs[bi][kk + koff    ][wc * 32 + 16 + l15];
            b1.y = Bs[bi][kk + koff + 1][wc * 32 + 16 + l15];
            acc00 = __builtin_amdgcn_wmma_f32_16x16x4_f32(false, a0, false, b0, (short)0, acc00, false, false);
            acc01 = __builtin_amdgcn_wmma_f32_16x16x4_f32(false, a0, false, b1, (short)0, acc01, false, false);
            acc10 = __builtin_amdgcn_wmma_f32_16x16x4_f32(false, a1, false, b0, (short)0, acc10, false, false);
            acc11 = __builtin_amdgcn_wmma_f32_16x16x4_f32(false, a1, false, b1, (short)0, acc11, false, false);
        }
    };

#if HAVE_ASYNC
    // Out-of-range A rows (last block only): zero-fill once; async copies skip
    // these rows every chunk, so the zeros persist for the whole k-loop.
    if (row0 + 64 > M) {
        #pragma unroll
        for (int p = 0; p < 2; ++p) {
            const int rr = r + p * 32;
            if (row0 + rr >= M) {
                float4 z; z.x = z.y = z.z = z.w = 0.f;
                *(float4*)&As[0][rr][q] = z;
                *(float4*)&As[1][rr][q] = z;
            }
        }
    }

    auto issue = [&](int kb, int bi) {
        const float* Ap; int stride, kloc;
        segsel(kb, Ap, stride, kloc);
        #pragma unroll
        for (int p = 0; p < 2; ++p) {               // A: 64x32
            const int rr = r + p * 32;
            const int grow = row0 + rr;
            if (grow < M)
                async_b128(&Ap[(size_t)grow * stride + kloc + q], &As[bi][rr][q]);
        }
        #pragma unroll
        for (int p = 0; p < 4; ++p) {               // B: 32x128 (never OOB)
            const int idx = tid + p * 256;
            const int brw = idx >> 5;
            const int bc  = (idx & 31) * 4;
            async_b128(&B[(size_t)(kb + brw) * 128 + bc], &Bs[bi][brw][bc]);
        }
    };

    const int nch = Ktot >> 5;
    issue(0, 0);
    for (int ci = 0; ci < nch; ++ci) {
        __builtin_amdgcn_s_wait_asynccnt(0);   // my chunk-ci loads landed
        __syncthreads();                       // everyone's landed; prev buf free
        if (ci + 1 < nch) issue((ci + 1) << 5, (ci + 1) & 1);
        compute(ci & 1);                       // overlaps next chunk's streaming
    }
#else
    for (int kb = 0; kb < Ktot; kb += 32) {
        const float* Ap; int stride, kloc;
        segsel(kb, Ap, stride, kloc);
        // Batch all global loads into registers first -> one s_wait_loadcnt.
        float4 ra[2], rb[4];
        #pragma unroll
        for (int p = 0; p < 2; ++p) {
            const int grow = row0 + r + p * 32;
            ra[p].x = ra[p].y = ra[p].z = ra[p].w = 0.f;
            if (grow < M)
                ra[p] = *(const float4*)&Ap[(size_t)grow * stride + kloc + q];
        }
        #pragma unroll
        for (int p = 0; p < 4; ++p) {
            const int idx = tid + p * 256;
            rb[p] = *(const float4*)&B[(size_t)(kb + (idx >> 5)) * 128 + (idx & 31) * 4];
        }
        __syncthreads();
        #pragma unroll
        for (int p = 0; p < 2; ++p)
            *(float4*)&As[0][r + p * 32][q] = ra[p];
        #pragma unroll
        for (int p = 0; p < 4; ++p) {
            const int idx = tid + p * 256;
            *(float4*)&Bs[0][idx >> 5][(idx & 31) * 4] = rb[p];
        }
        __syncthreads();
        compute(0);
    }
#endif

    // Epilogue. C/D layout: VGPR v, lanes0-15 -> M=v ; lanes16-31 -> M=v+8.
    const int rquad = (lane >> 4) * 8;
    auto store_tile = [&](const v8f& a, int mbase, int nbase) {
        const int col = nbase + l15;
        #pragma unroll
        for (int v = 0; v < 8; ++v) {
            const int row = row0 + mbase + rquad + v;
            if (row < M) {
                float rv = a[v];
                if (bias) rv += bias[col];
                if (doLrelu) rv = (rv > 0.f) ? rv : rv * SLOPE;
                if (Res) rv += Res[(size_t)row * rstride + roff + col];
                Out[(size_t)row * ostride + ooff + col] = rv;
            }
        }
    };
    store_tile(acc00, wr * 32,      wc * 32);
    store_tile(acc01, wr * 32,      wc * 32 + 16);
    store_tile(acc10, wr * 32 + 16, wc * 32);
    store_tile(acc11, wr * 32 + 16, wc * 32 + 16);
}

// ---------------------------------------------------------------------------
// Neighbor max aggregation + EdgeConv combine:
//   H[i,c] = lrelu(Tself[i,c] - Tnbr[i,c] + bias[c] + max_j Tnbr[src[i,j],c])
// 2 nodes per 256-thread block; gathers are 512B-contiguous rows, L2-resident.
// ---------------------------------------------------------------------------
__global__ __launch_bounds__(256)
void edge_max_kernel(const float* __restrict__ Tself,
                     const float* __restrict__ Tnbr,
                     const float* __restrict__ bias,
                     const int* __restrict__ src,   // [N*K], grouped by dst
                     float* __restrict__ H, int nNodes)
{
    __shared__ int sidx[2][KNBR];
    const int tid  = threadIdx.x;
    const int half = tid >> 7;       // node within block
    const int c    = tid & 127;      // channel
    const int node = blockIdx.x * 2 + half;
    if (c < KNBR && node < nNodes)
        sidx[half][c] = src[node * KNBR + c];
    __syncthreads();
    if (node >= nNodes) return;

    float m = -__builtin_inff();
    #pragma unroll
    for (int j = 0; j < KNBR; ++j) {
        const int s = sidx[half][j];
        m = fmaxf(m, Tnbr[(size_t)s * CD + c]);
    }
    const float z = Tself[(size_t)node * CD + c]
                  - Tnbr [(size_t)node * CD + c]
                  + bias[c] + m;
    H[(size_t)node * CD + c] = (z > 0.f) ? z : z * SLOPE;
}

// ---------------------------------------------------------------------------
extern "C" void kernel_launch(void* const* d_in, const int* in_sizes, int n_in,
                              void* d_out, int out_size, void* d_ws, size_t ws_size,
                              hipStream_t stream)
{
    (void)in_sizes; (void)n_in; (void)out_size; (void)ws_size;

    const float* x   = (const float*)d_in[0];             // [N,256]
    const int*   ei1 = (const int*)d_in[1];               // [2, N*K]; row0=src
    const int*   ei2 = (const int*)d_in[2];
    const float* Wb  = (const float*)d_in[3];             // [256,128]
    const float* bb  = (const float*)d_in[4];             // [128]
    // per branch (pytree order): Ws1,Wn1,bs1, Ws2,Wn2,bs2, Wd,bd
    float* out = (float*)d_out;

    float* ws    = (float*)d_ws;
    float* xb    = ws;                                    // [N,128]
    float* h1    = xb    + (size_t)NNODES * CD;
    float* h2    = h1    + (size_t)NNODES * CD;
    float* Tself = h2    + (size_t)NNODES * CD;
    float* Tnbr  = Tself + (size_t)NNODES * CD;

    const dim3 blk(256);
    const dim3 ggrid((NNODES + 63) / 64);
    const dim3 agrid((NNODES + 1) / 2);

    // Bottleneck: xb = lrelu(x @ Wb + bb)
    gemm128_wmma<<<ggrid, blk, 0, stream>>>(x, CIN, nullptr, 0, nullptr, 0,
                                            Wb, bb, xb, CD, 0,
                                            nullptr, 0, 0, NNODES, 1);

    for (int br = 0; br < 2; ++br) {
        const float* Ws1 = (const float*)d_in[5 + br * 8 + 0];
        const float* Wn1 = (const float*)d_in[5 + br * 8 + 1];
        const float* bs1 = (const float*)d_in[5 + br * 8 + 2];
        const float* Ws2 = (const float*)d_in[5 + br * 8 + 3];
        const float* Wn2 = (const float*)d_in[5 + br * 8 + 4];
        const float* bs2 = (const float*)d_in[5 + br * 8 + 5];
        const float* Wd  = (const float*)d_in[5 + br * 8 + 6];
        const float* bd  = (const float*)d_in[5 + br * 8 + 7];
        const int*   src = (br == 0) ? ei1 : ei2;         // row 0 = src

        // Block 1 (K=128): Tself = xb@Ws1, Tnbr = xb@Wn1, then gather-max
        gemm128_wmma<<<ggrid, blk, 0, stream>>>(xb, CD, nullptr, 0, nullptr, 0,
                                                Ws1, nullptr, Tself, CD, 0,
                                                nullptr, 0, 0, NNODES, 0);
        gemm128_wmma<<<ggrid, blk, 0, stream>>>(xb, CD, nullptr, 0, nullptr, 0,
                                                Wn1, nullptr, Tnbr, CD, 0,
                                                nullptr, 0, 0, NNODES, 0);
        edge_max_kernel<<<agrid, blk, 0, stream>>>(Tself, Tnbr, bs1, src, h1, NNODES);

        // Block 2 (K=256 = [xb|h1]):
        gemm128_wmma<<<ggrid, blk, 0, stream>>>(xb, CD, h1, CD, nullptr, 0,
                                                Ws2, nullptr, Tself, CD, 0,
                                                nullptr, 0, 0, NNODES, 0);
        gemm128_wmma<<<ggrid, blk, 0, stream>>>(xb, CD, h1, CD, nullptr, 0,
                                                Wn2, nullptr, Tnbr, CD, 0,
                                                nullptr, 0, 0, NNODES, 0);
        edge_max_kernel<<<agrid, blk, 0, stream>>>(Tself, Tnbr, bs2, src, h2, NNODES);

        // lin_down (K=384 = [xb|h1|h2]) fused with residual, direct to d_out half
        gemm128_wmma<<<ggrid, blk, 0, stream>>>(xb, CD, h1, CD, h2, CD,
                                                Wd, bd, out, CIN, br * CD,
                                                x, CIN, br * CD, NNODES, 0);
    }
}